// KanLayer_70875550319074
// MI455X (gfx1250) — compile-verified
//
#include <hip/hip_runtime.h>

typedef __attribute__((ext_vector_type(2))) float v2f;
typedef __attribute__((ext_vector_type(8))) float v8f;
typedef int v4i __attribute__((__vector_size__(4 * sizeof(int))));
typedef __attribute__((address_space(1))) v4i* global_v4i_ptr;
typedef __attribute__((address_space(3))) v4i* lds_v4i_ptr;

#define OUT_DIM   4096
#define N_COEF    61                      // = 64 knots - (degree 2 + 1)
#define I_TILE    64                      // i's per block
#define O_TILE    64                      // outputs per block (4 wave-groups x 16)
#define CHUNK_FLOATS (O_TILE * N_COEF)    // 3904
#define CHUNK_VEC4   (CHUNK_FLOATS / 4)   // 976
#define TAIL_THREADS (CHUNK_VEC4 - 768)   // 208 threads active in 4th segment
#define BUF_FLOATS   (CHUNK_FLOATS + 4)   // 3908: 4-float zeroed pad for k=61..63 reads
#define N_BUF     3                       // prefetch depth 2
#define BS_STRIDE    65                   // conflict-free LDS banking for basis rows
#define C_I_STRIDE   (OUT_DIM * N_COEF)   // 249856 floats per i

__device__ __forceinline__ float knot(int j) {
    // t = linspace(-1, 1, 64)
    return -1.0f + (2.0f / 63.0f) * (float)j;
}

// ---- CDNA5 async global->LDS copy (ASYNCcnt-tracked, no VGPR data) ----------
__device__ __forceinline__ void async_load_b128(const float* g, float* l) {
#if __has_builtin(__builtin_amdgcn_global_load_async_to_lds_b128)
    __builtin_amdgcn_global_load_async_to_lds_b128(
        (global_v4i_ptr)g, (lds_v4i_ptr)l, 0, 0);
#else
    unsigned lofs = (unsigned)(unsigned long long)l;  // LDS addr = generic[31:0]
    asm volatile("global_load_async_to_lds_b128 %0, %1, off"
                 :: "v"(lofs), "v"(g) : "memory");
#endif
}

__device__ __forceinline__ void wait_async0() {
#if __has_builtin(__builtin_amdgcn_s_wait_asynccnt)
    __builtin_amdgcn_s_wait_asynccnt(0);
#else
    asm volatile("s_wait_asynccnt 0x0" ::: "memory");
#endif
}

__device__ __forceinline__ void wait_async4() {   // oldest chunk done, next may fly
#if __has_builtin(__builtin_amdgcn_s_wait_asynccnt)
    __builtin_amdgcn_s_wait_asynccnt(4);
#else
    asm volatile("s_wait_asynccnt 0x4" ::: "memory");
#endif
}

__global__ void __launch_bounds__(256) kan_zero_kernel(float* out, int n) {
    int i = blockIdx.x * 256 + threadIdx.x;
    if (i < n) out[i] = 0.0f;
}

__global__ void __launch_bounds__(256) kan_wmma_kernel(const float* __restrict__ x,
                                                       const float* __restrict__ c,
                                                       float* __restrict__ out) {
    __shared__ float buf[N_BUF][BUF_FLOATS];        // 3 x 15.6 KB rotating
    __shared__ float bs[I_TILE * BS_STRIDE];        // basis, k padded to 0

    const int tid = threadIdx.x;
    const int bo  = blockIdx.x & 63;    // 64 output tiles
    const int bi  = blockIdx.x >> 6;    // 32 i chunks
    const int o0  = bo * O_TILE;
    const int i0  = bi * I_TILE;

    // ---------------- Phase A: Cox-de Boor basis, in-place, 1 thread per i ---
    if (tid < I_TILE) {
        float xv = x[i0 + tid];
        float* row = &bs[tid * BS_STRIDE];
        for (int j = 0; j < 63; ++j)
            row[j] = (knot(j) <= xv && xv < knot(j + 1)) ? 1.0f : 0.0f;
        row[63] = 0.0f;
        for (int j = 0; j < 62; ++j) {          // p = 1 (forward in-place is safe)
            float l = (xv - knot(j)) / (knot(j + 1) - knot(j));
            float r = (knot(j + 2) - xv) / (knot(j + 2) - knot(j + 1));
            row[j] = l * row[j] + r * row[j + 1];
        }
        for (int j = 0; j < 61; ++j) {          // p = 2
            float l = (xv - knot(j)) / (knot(j + 2) - knot(j));
            float r = (knot(j + 3) - xv) / (knot(j + 3) - knot(j + 1));
            row[j] = l * row[j] + r * row[j + 1];
        }
        row[61] = 0.0f; row[62] = 0.0f; row[63] = 0.0f;  // kill WMMA K-padding
    } else if (tid < I_TILE + 4 * N_BUF) {
        int t = tid - I_TILE;                   // zero the buffer tail pads
        buf[t >> 2][CHUNK_FLOATS + (t & 3)] = 0.0f;
    }

    // ---------------- Phase B: async-stream c, WMMA-accumulate ---------------
    const int lane = tid & 31;
    const int wave = tid >> 5;
    const int hv   = lane >> 4;                 // 0 => K{+0,+1}, 1 => K{+2,+3}
    const int m    = lane & 15;
    const int ow   = (wave & 3) * 16;           // 16-output group
    const int kb   = (wave >> 2) * 32;          // k half: 0..31 or 32..63
    const int a_base  = (ow + m) * N_COEF + kb + hv * 2;
    const int b_base0 = kb + hv * 2;

    // per-thread global sources (4 float4 segments; only segment 3 is partial)
    const float* gptr0; const float* gptr1; const float* gptr2; const float* gptr3;
    {
        const long long base0 = (long long)i0 * C_I_STRIDE + (long long)o0 * N_COEF;
        gptr0 = c + base0 + 4 * (tid);
        gptr1 = c + base0 + 4 * (tid + 256);
        gptr2 = c + base0 + 4 * (tid + 512);
        gptr3 = c + base0 + 4 * (tid + 768);
    }
    const int dof0 = 4 * tid, dof1 = 4 * (tid + 256),
              dof2 = 4 * (tid + 512), dof3 = 4 * (tid + 768);
    const bool tail = (tid < TAIL_THREADS);

    auto issue_chunk = [&](float* dst) {
        async_load_b128(gptr0, dst + dof0);
        async_load_b128(gptr1, dst + dof1);
        async_load_b128(gptr2, dst + dof2);
        if (tail) async_load_b128(gptr3, dst + dof3);
        gptr0 += C_I_STRIDE; gptr1 += C_I_STRIDE;
        gptr2 += C_I_STRIDE; gptr3 += C_I_STRIDE;
    };

    __syncthreads();   // basis + pads visible

    // prologue: chunks 0 and 1 in flight
    issue_chunk(&buf[0][0]);
    issue_chunk(&buf[1][0]);

    float* cb = &buf[0][0];     // holds chunk i
    float* nb = &buf[1][0];     // holds chunk i+1 (in flight)
    float* pb = &buf[2][0];     // free: target for chunk i+2

    v8f acc = {0.f, 0.f, 0.f, 0.f, 0.f, 0.f, 0.f, 0.f};

    for (int i = 0; i < I_TILE; ++i) {
        if (i + 1 < I_TILE) wait_async4();   // oldest (chunk i) landed, i+1 may fly
        else                wait_async0();   // last chunk: wait fully
        __syncthreads();    // everyone's chunk i landed; pb free for reuse

        if (i + 2 < I_TILE)
            issue_chunk(pb);     // overlaps the WMMA phase below

        const int bb = i * BS_STRIDE + b_base0;
        #pragma unroll
        for (int k0 = 0; k0 < 32; k0 += 4) {
            v2f a, b;
            // A: 16x4 f32 — lanes 0-15 K=k0,k0+1; lanes 16-31 K=k0+2,k0+3
            a[0] = cb[a_base + k0];
            a[1] = cb[a_base + k0 + 1];
            // B: basis broadcast across all 16 N columns
            b[0] = bs[bb + k0];
            b[1] = bs[bb + k0 + 1];
            acc = __builtin_amdgcn_wmma_f32_16x16x4_f32(
                false, a, false, b, (short)0, acc, false, false);
        }

        float* t = cb; cb = nb; nb = pb; pb = t;   // rotate buffers
    }

    // every D column identical: lane 0 -> M 0..7, lane 16 -> M 8..15
    if (m == 0) {
        int base = o0 + ow + hv * 8;
        #pragma unroll
        for (int j = 0; j < 8; ++j)
            atomicAdd(&out[base + j], acc[j]);
    }
}

extern "C" void kernel_launch(void* const* d_in, const int* in_sizes, int n_in,
                              void* d_out, int out_size, void* d_ws, size_t ws_size,
                              hipStream_t stream) {
    const float* x = (const float*)d_in[0];   // (2048,)
    const float* c = (const float*)d_in[1];   // (2048, 4096, 61)
    float* out = (float*)d_out;               // (4096,)

    kan_zero_kernel<<<(OUT_DIM + 255) / 256, 256, 0, stream>>>(out, OUT_DIM);

    // 64 o-tiles x 32 i-chunks
    kan_wmma_kernel<<<dim3(64 * 32), dim3(256), 0, stream>>>(x, c, out);
}